// NodeEncoder_16535624090331
// MI455X (gfx1250) — compile-verified
//
#include <hip/hip_runtime.h>

#define DIM 128
#define KPSTRIDE 288  // padded K-pair row stride (floats); 288 % 64 == 32 -> conflict-free halves

typedef __attribute__((ext_vector_type(2))) float v2f;
typedef __attribute__((ext_vector_type(8))) float v8f;

// ---------------------------------------------------------------------------
// Degree / normalization precompute
// ---------------------------------------------------------------------------
__global__ void deg_init_kernel(float* __restrict__ deg, int n) {
  int i = blockIdx.x * blockDim.x + threadIdx.x;
  if (i < n) deg[i] = 1.0f;  // self-loop
}

__global__ void deg_accum_kernel(const int* __restrict__ dst, float* __restrict__ deg, int ne) {
  int e = blockIdx.x * blockDim.x + threadIdx.x;
  if (e < ne) atomicAdd(&deg[dst[e]], 1.0f);
}

__global__ void deg_rsqrt_kernel(float* __restrict__ deg, int n) {
  int i = blockIdx.x * blockDim.x + threadIdx.x;
  if (i < n) deg[i] = rsqrtf(deg[i]);
}

// ---------------------------------------------------------------------------
// GEMM: M[n x 128] = H[n x 128] * W[128 x 128] via V_WMMA_F32_16X16X4_F32.
// 256 threads = 8 waves; each wave owns a 16-row slab and all 8 N-tiles
// (8 x v8f accumulators).
// W is staged in LDS in K-pair-interleaved layout so each lane's B fragment
// {W[kb][n], W[kb+1][n]} is one contiguous ds_load_b64 (no shuffle movs),
// with a 288-float K-pair stride for bank-conflict-free access.
// Uniform control flow up to the WMMAs => EXEC all-ones as required.
// ---------------------------------------------------------------------------
__global__ void gemm_wmma_kernel(const float* __restrict__ H,
                                 const float* __restrict__ W,
                                 float* __restrict__ M, int nrows) {
  __shared__ float ldsB[(DIM / 2) * KPSTRIDE];  // 64 * 288 * 4B = 72 KB
  const int t = threadIdx.x;

  // Stage W: coalesced global reads, interleaved LDS writes.
  for (int i = t; i < DIM * DIM; i += 256) {
    const int k = i >> 7;    // W row (K index)
    const int nn = i & 127;  // W col (N index)
    ldsB[(k >> 1) * KPSTRIDE + nn * 2 + (k & 1)] = W[i];
  }
  __syncthreads();

  const int lane = t & 31;
  const int wave = t >> 5;
  const int lo = lane & 15;  // M (for A) / N (for B,C,D) position
  const int hi = lane >> 4;  // K-pair select for A/B, row-half for C/D
  const int row0 = blockIdx.x * 128 + wave * 16;

  // A-fragment source row (clamped; out-of-range rows computed but never stored)
  int aRow = row0 + lo;
  if (aRow > nrows - 1) aRow = nrows - 1;
  const float* __restrict__ arow = H + (size_t)aRow * DIM;

  v8f acc[8] = {};

#pragma unroll 4
  for (int k0 = 0; k0 < DIM; k0 += 4) {
    const int kb = k0 + 2 * hi;  // even
    v2f a;
    a.x = arow[kb + 0];
    a.y = arow[kb + 1];
    const float* __restrict__ bb = &ldsB[(kb >> 1) * KPSTRIDE + lo * 2];
#pragma unroll
    for (int nt = 0; nt < 8; ++nt) {
      const v2f bf = *(const v2f*)&bb[nt * 32];  // ds_load_b64, imm offset nt*128B
      // (neg_a, A, neg_b, B, c_mod, C, reuse_a, reuse_b)
      acc[nt] = __builtin_amdgcn_wmma_f32_16x16x4_f32(
          false, a, false, bf, (short)0, acc[nt], false, false);
    }
  }

#pragma unroll
  for (int nt = 0; nt < 8; ++nt) {
#pragma unroll
    for (int r = 0; r < 8; ++r) {
      const int row = row0 + r + 8 * hi;  // C/D layout: vgpr r -> M = r + 8*hi
      if (row < nrows) M[(size_t)row * DIM + nt * 16 + lo] = acc[nt][r];
    }
  }
}

// ---------------------------------------------------------------------------
// out[v] = b + dinv[v]^2 * m[v]   (self-loop fused, no atomics, full overwrite)
// one wave per node, float4 per lane
// ---------------------------------------------------------------------------
__global__ void agg_init_kernel(const float* __restrict__ m, const float* __restrict__ dinv,
                                const float* __restrict__ bias, float* __restrict__ out, int n) {
  int t = blockIdx.x * blockDim.x + threadIdx.x;
  int node = t >> 5;
  int c = (t & 31) * 4;
  if (node < n) {
    float s = dinv[node];
    s *= s;
    const float4 mv = *(const float4*)&m[(size_t)node * DIM + c];
    const float4 bv = *(const float4*)&bias[c];
    float4 o;
    o.x = bv.x + s * mv.x;
    o.y = bv.y + s * mv.y;
    o.z = bv.z + s * mv.z;
    o.w = bv.w + s * mv.w;
    *(float4*)&out[(size_t)node * DIM + c] = o;
  }
}

// ---------------------------------------------------------------------------
// out[dst] += dinv[src]*dinv[dst] * m[src]  -- one wave per edge
// ---------------------------------------------------------------------------
__global__ void agg_edge_kernel(const float* __restrict__ m, const float* __restrict__ dinv,
                                const int* __restrict__ src, const int* __restrict__ dst,
                                float* __restrict__ out, int ne) {
  int t = blockIdx.x * blockDim.x + threadIdx.x;
  int e = t >> 5;
  int c = (t & 31) * 4;
  if (e < ne) {
    const int s = src[e];
    const int d = dst[e];
    const float w = dinv[s] * dinv[d];
    const float4 mv = *(const float4*)&m[(size_t)s * DIM + c];
    float* o = &out[(size_t)d * DIM + c];
    atomicAdd(o + 0, w * mv.x);
    atomicAdd(o + 1, w * mv.y);
    atomicAdd(o + 2, w * mv.z);
    atomicAdd(o + 3, w * mv.w);
  }
}

// ---------------------------------------------------------------------------
// BatchNorm statistics + apply
// ---------------------------------------------------------------------------
__global__ void stats_zero_kernel(float* __restrict__ sums) {
  sums[threadIdx.x] = 0.0f;  // 256 threads: sums[0..127], sumsq[128..255]
}

__global__ void stats_accum_kernel(const float* __restrict__ h, float* __restrict__ sums, int n) {
  const int d = threadIdx.x;  // 128 threads, column-coalesced rows
  const int r0 = blockIdx.x * 128;
  int r1 = r0 + 128;
  if (r1 > n) r1 = n;
  float s = 0.0f, s2 = 0.0f;
  for (int r = r0; r < r1; ++r) {
    const float v = h[(size_t)r * DIM + d];
    s += v;
    s2 += v * v;
  }
  atomicAdd(&sums[d], s);
  atomicAdd(&sums[DIM + d], s2);
}

__global__ void stats_final_kernel(const float* __restrict__ sums,
                                   const float* __restrict__ gamma,
                                   const float* __restrict__ beta,
                                   float* __restrict__ scsh, float inv_n) {
  const int d = threadIdx.x;  // 128 threads
  const float mu = sums[d] * inv_n;
  const float var = sums[DIM + d] * inv_n - mu * mu;  // biased variance
  const float rs = rsqrtf(var + 1e-5f);
  const float sc = gamma[d] * rs;
  scsh[d] = sc;
  scsh[DIM + d] = beta[d] - mu * sc;
}

__global__ void bn_relu_kernel(float* __restrict__ h, const float* __restrict__ scsh, size_t total) {
  size_t idx = (size_t)blockIdx.x * blockDim.x + threadIdx.x;
  if (idx < total) {
    const int d = (int)(idx & (DIM - 1));
    const float v = h[idx] * scsh[d] + scsh[DIM + d];
    h[idx] = v > 0.0f ? v : 0.0f;
  }
}

// ---------------------------------------------------------------------------
extern "C" void kernel_launch(void* const* d_in, const int* in_sizes, int n_in,
                              void* d_out, int out_size, void* d_ws, size_t ws_size,
                              hipStream_t stream) {
  (void)n_in; (void)out_size; (void)ws_size;

  const float* x     = (const float*)d_in[0];
  const int*   edge  = (const int*)d_in[1];
  const float* W     = (const float*)d_in[2];
  const float* b     = (const float*)d_in[3];
  const float* gamma = (const float*)d_in[4];
  const float* beta  = (const float*)d_in[5];

  const int n  = in_sizes[0] / DIM;          // 50000
  const int ne = in_sizes[1] / 2;            // 800000
  const int L  = in_sizes[2] / (DIM * DIM);  // 3

  const int* srcIdx = edge;       // edge_index[0]
  const int* dstIdx = edge + ne;  // edge_index[1]

  float* ws   = (float*)d_ws;
  float* mbuf = ws;                        // N*D
  float* hbuf = ws + (size_t)n * DIM;      // N*D
  float* dinv = ws + 2 * (size_t)n * DIM;  // N
  float* sums = dinv + n;                  // 256
  float* scsh = sums + 256;                // 256

  const dim3 blk(256);

  deg_init_kernel<<<dim3((n + 255) / 256), blk, 0, stream>>>(dinv, n);
  deg_accum_kernel<<<dim3((ne + 255) / 256), blk, 0, stream>>>(dstIdx, dinv, ne);
  deg_rsqrt_kernel<<<dim3((n + 255) / 256), blk, 0, stream>>>(dinv, n);

  const float* hin = x;
  for (int l = 0; l < L; ++l) {
    gemm_wmma_kernel<<<dim3((n + 127) / 128), blk, 0, stream>>>(
        hin, W + (size_t)l * DIM * DIM, mbuf, n);

    float* hout = (l == L - 1) ? (float*)d_out : hbuf;

    const unsigned initBlocks = (unsigned)(((size_t)n * 32 + 255) / 256);
    agg_init_kernel<<<dim3(initBlocks), blk, 0, stream>>>(mbuf, dinv, b + (size_t)l * DIM, hout, n);

    const unsigned edgeBlocks = (unsigned)(((size_t)ne * 32 + 255) / 256);
    agg_edge_kernel<<<dim3(edgeBlocks), blk, 0, stream>>>(mbuf, dinv, srcIdx, dstIdx, hout, ne);

    stats_zero_kernel<<<dim3(1), blk, 0, stream>>>(sums);
    stats_accum_kernel<<<dim3((n + 127) / 128), dim3(128), 0, stream>>>(hout, sums, n);
    stats_final_kernel<<<dim3(1), dim3(128), 0, stream>>>(
        sums, gamma + (size_t)l * DIM, beta + (size_t)l * DIM, scsh, 1.0f / (float)n);

    const size_t total = (size_t)n * DIM;
    bn_relu_kernel<<<dim3((unsigned)((total + 255) / 256)), blk, 0, stream>>>(hout, scsh, total);

    hin = hout;
  }
}